// GNN_88648124990247
// MI455X (gfx1250) — compile-verified
//
#include <hip/hip_runtime.h>

#define NODES   100000
#define EDGES   1200000
#define GRAPHS  1000
#define HIDDIM  64
#define NRELS   3

typedef __attribute__((ext_vector_type(2))) float v2f;
typedef __attribute__((ext_vector_type(8))) float v8f;

__device__ __forceinline__ void atomic_add_f(float* p, float v) {
  __hip_atomic_fetch_add(p, v, __ATOMIC_RELAXED, __HIP_MEMORY_SCOPE_AGENT);
}

// ---------------------------------------------------------------- utilities
__global__ __launch_bounds__(256) void zero_kernel(float* __restrict__ p, int n) {
  int i = blockIdx.x * 256 + threadIdx.x;
  if (i < n) p[i] = 0.0f;
}

// count in-degree per relation: cnt[r*N + dst] += 1
__global__ __launch_bounds__(256) void count_kernel(const int* __restrict__ e0,
                                                    const int* __restrict__ e1,
                                                    const int* __restrict__ e2,
                                                    float* __restrict__ cnt) {
  int t = blockIdx.x * 256 + threadIdx.x;
  if (t >= 3 * EDGES) return;
  int r = t / EDGES, e = t - r * EDGES;
  const int* ei = (r == 0) ? e0 : ((r == 1) ? e1 : e2);
  int dst = ei[EDGES + e];                      // row 1 = dst
  atomic_add_f(&cnt[(size_t)r * NODES + dst], 1.0f);
}

__global__ __launch_bounds__(256) void invert_kernel(float* __restrict__ c, int n) {
  int i = blockIdx.x * 256 + threadIdx.x;
  if (i < n) c[i] = 1.0f / fmaxf(c[i], 1.0f);
}

// Wsum = sum_r Wr[r] (64x64), bsum = sum_r bl[r] (64)
__global__ __launch_bounds__(256) void wsum_kernel(const float* __restrict__ Wr,
                                                   const float* __restrict__ bl,
                                                   float* __restrict__ Wsum,
                                                   float* __restrict__ bsum) {
  int tid = threadIdx.x;
  for (int i = tid; i < HIDDIM * HIDDIM; i += 256)
    Wsum[i] = Wr[i] + Wr[HIDDIM * HIDDIM + i] + Wr[2 * HIDDIM * HIDDIM + i];
  if (tid < HIDDIM)
    bsum[tid] = bl[tid] + bl[HIDDIM + tid] + bl[2 * HIDDIM + tid];
}

// ------------------------------------------------- WMMA f32 16x16x4 helpers
// A 16x4: lanes 0-15 -> M, VGPR j / lane-half h -> K = 2h + j
// B 4x16: lanes 0-15 -> N, VGPR j / lane-half h -> K = 2h + j
// C 16x16: VGPR i -> M = i + 8*(lane>>4), N = lane&15

// x0 = relu(concat(shape_emb[sid], color_emb[cid]) @ W0^T + b0)
__global__ __launch_bounds__(256) void embed_mlp_kernel(
    const int* __restrict__ sids, const int* __restrict__ cids,
    const float* __restrict__ semb, const float* __restrict__ cemb,
    const float* __restrict__ W0, const float* __restrict__ b0,
    float* __restrict__ Y) {
  __shared__ float WT[64 * 65];
  __shared__ float SE[32 * 32];
  __shared__ float CE[16 * 32];
  __shared__ float bsh[64];
  int tid = threadIdx.x;
  for (int i = tid; i < 4096; i += 256)
    WT[(i >> 6) * 65 + (i & 63)] = W0[(size_t)(i & 63) * 64 + (i >> 6)]; // WT[kin][nout]
  for (int i = tid; i < 1024; i += 256) SE[i] = semb[i];
  for (int i = tid; i < 512; i += 256) CE[i] = cemb[i];
  if (tid < 64) bsh[tid] = b0[tid];
  __syncthreads();

  int lane = tid & 31, wave = tid >> 5;
  int m16 = lane & 15, khalf = lane >> 4;
  int row0 = (blockIdx.x * 8 + wave) * 16;
  int rowA = row0 + m16;
  bool inA = rowA < NODES;
  int sid = inA ? sids[rowA] : 0;
  int cid = inA ? cids[rowA] : 0;

  v2f A[16];
#pragma unroll
  for (int k = 0; k < 16; ++k) {
    int f = k * 4 + khalf * 2;
    float ax = (f < 32) ? SE[sid * 32 + f] : CE[cid * 32 + (f - 32)];
    float ay = ((f + 1) < 32) ? SE[sid * 32 + f + 1] : CE[cid * 32 + (f + 1 - 32)];
    A[k].x = inA ? ax : 0.0f;
    A[k].y = inA ? ay : 0.0f;
  }

#pragma unroll
  for (int nt = 0; nt < 4; ++nt) {
    int n = nt * 16 + m16;
    v8f C = {};
#pragma unroll
    for (int k = 0; k < 16; ++k) {
      int kr = k * 4 + khalf * 2;
      v2f B;
      B.x = WT[kr * 65 + n];
      B.y = WT[(kr + 1) * 65 + n];
      C = __builtin_amdgcn_wmma_f32_16x16x4_f32(false, A[k], false, B,
                                                (short)0, C, false, false);
    }
#pragma unroll
    for (int i = 0; i < 8; ++i) {
      int row = row0 + i + khalf * 8;
      if (row < NODES)
        Y[(size_t)row * 64 + n] = fmaxf(C[i] + bsh[n], 0.0f);
    }
  }
}

// Y (+)= (X * rowscale?) @ W^T (+ bias) (relu?)   mode bit0=accumulate, bit1=relu
__global__ __launch_bounds__(256) void matmul64_kernel(
    const float* __restrict__ X, const float* __restrict__ rowscale,
    const float* __restrict__ W, const float* __restrict__ bias,
    float* __restrict__ Y, int mode) {
  __shared__ float WT[64 * 65];
  __shared__ float bsh[64];
  int tid = threadIdx.x;
  for (int i = tid; i < 4096; i += 256)
    WT[(i >> 6) * 65 + (i & 63)] = W[(size_t)(i & 63) * 64 + (i >> 6)];
  if (tid < 64) bsh[tid] = bias ? bias[tid] : 0.0f;
  __syncthreads();

  int lane = tid & 31, wave = tid >> 5;
  int m16 = lane & 15, khalf = lane >> 4;
  int row0 = (blockIdx.x * 8 + wave) * 16;
  int rowA = row0 + m16;
  bool inA = rowA < NODES;
  float sc = (inA && rowscale) ? rowscale[rowA] : 1.0f;
  const float* xrow = X + (size_t)(inA ? rowA : 0) * 64;

  v2f A[16];
#pragma unroll
  for (int k = 0; k < 16; ++k) {
    int f = k * 4 + khalf * 2;
    A[k].x = inA ? xrow[f] * sc : 0.0f;
    A[k].y = inA ? xrow[f + 1] * sc : 0.0f;
  }

#pragma unroll
  for (int nt = 0; nt < 4; ++nt) {
    int n = nt * 16 + m16;
    v8f C = {};
#pragma unroll
    for (int k = 0; k < 16; ++k) {
      int kr = k * 4 + khalf * 2;
      v2f B;
      B.x = WT[kr * 65 + n];
      B.y = WT[(kr + 1) * 65 + n];
      C = __builtin_amdgcn_wmma_f32_16x16x4_f32(false, A[k], false, B,
                                                (short)0, C, false, false);
    }
#pragma unroll
    for (int i = 0; i < 8; ++i) {
      int row = row0 + i + khalf * 8;
      if (row < NODES) {
        size_t idx = (size_t)row * 64 + n;
        float v = C[i] + bsh[n];
        if (mode & 1) v += Y[idx];
        if (mode & 2) v = fmaxf(v, 0.0f);
        Y[idx] = v;
      }
    }
  }
}

// agg[dst] += X[src]  (4 threads per edge, 16 floats each, B128 reads)
__global__ __launch_bounds__(256) void scatter_kernel(const float* __restrict__ X,
                                                      const int* __restrict__ ei,
                                                      float* __restrict__ agg) {
  int t = blockIdx.x * 256 + threadIdx.x;
  int e = t >> 2;
  if (e >= EDGES) return;
  int part = (t & 3) * 16;
  int src = ei[e];
  int dst = ei[EDGES + e];
  const float4* xs = (const float4*)(X + (size_t)src * 64 + part);
  float* ad = agg + (size_t)dst * 64 + part;
#pragma unroll
  for (int q = 0; q < 4; ++q) {
    float4 v = xs[q];
    atomic_add_f(ad + q * 4 + 0, v.x);
    atomic_add_f(ad + q * 4 + 1, v.y);
    atomic_add_f(ad + q * 4 + 2, v.z);
    atomic_add_f(ad + q * 4 + 3, v.w);
  }
}

// pooled[batch[n]] += X[n]; gcnt[batch[n]] += 1
__global__ __launch_bounds__(256) void pool_kernel(const float* __restrict__ X,
                                                   const int* __restrict__ batch,
                                                   float* __restrict__ pooled,
                                                   float* __restrict__ gcnt) {
  int t = blockIdx.x * 256 + threadIdx.x;
  int n = t >> 2;
  if (n >= NODES) return;
  int part = (t & 3) * 16;
  int g = batch[n];
  if ((t & 3) == 0) atomic_add_f(&gcnt[g], 1.0f);
  const float* xr = X + (size_t)n * 64 + part;
  float* pr = pooled + (size_t)g * 64 + part;
#pragma unroll
  for (int j = 0; j < 16; ++j) atomic_add_f(pr + j, xr[j]);
}

// out[g,c] = mean_pooled[g] . Wout[c] + bout[c]
__global__ __launch_bounds__(256) void out_kernel(const float* __restrict__ pooled,
                                                  const float* __restrict__ gcnt,
                                                  const float* __restrict__ Wout,
                                                  const float* __restrict__ bout,
                                                  float* __restrict__ out) {
  int g = blockIdx.x * 256 + threadIdx.x;
  if (g >= GRAPHS) return;
  float inv = 1.0f / fmaxf(gcnt[g], 1.0f);
  float a0 = 0.0f, a1 = 0.0f;
  const float* p = pooled + (size_t)g * 64;
#pragma unroll
  for (int h = 0; h < 64; ++h) {
    float v = p[h] * inv;
    a0 += v * Wout[h];
    a1 += v * Wout[64 + h];
  }
  out[g * 2 + 0] = a0 + bout[0];
  out[g * 2 + 1] = a1 + bout[1];
}

// ---------------------------------------------------------------- launcher
extern "C" void kernel_launch(void* const* d_in, const int* in_sizes, int n_in,
                              void* d_out, int out_size, void* d_ws, size_t ws_size,
                              hipStream_t stream) {
  const int* shape_ids = (const int*)d_in[0];
  const int* color_ids = (const int*)d_in[1];
  const int* ei[NRELS] = {(const int*)d_in[2], (const int*)d_in[3], (const int*)d_in[4]};
  const int* batch     = (const int*)d_in[5];
  const float* semb = (const float*)d_in[6];
  const float* cemb = (const float*)d_in[7];
  const float* W0   = (const float*)d_in[8];
  const float* b0   = (const float*)d_in[9];
  const float* Wl[2] = {(const float*)d_in[10], (const float*)d_in[13]};
  const float* bl[2] = {(const float*)d_in[11], (const float*)d_in[14]};
  const float* Wr[2] = {(const float*)d_in[12], (const float*)d_in[15]};
  const float* Wout = (const float*)d_in[16];
  const float* bout = (const float*)d_in[17];
  float* out = (float*)d_out;

  float* ws     = (float*)d_ws;
  float* xA     = ws;                                     // N*64
  float* xB     = xA + (size_t)NODES * HIDDIM;            // N*64
  float* agg    = xB + (size_t)NODES * HIDDIM;            // N*64
  float* invcnt = agg + (size_t)NODES * HIDDIM;           // 3*N
  float* pooled = invcnt + 3 * (size_t)NODES;             // G*64
  float* gcnt   = pooled + (size_t)GRAPHS * HIDDIM;       // G (contiguous w/ pooled)
  float* Wsum[2], *bsum[2];
  Wsum[0] = gcnt + GRAPHS;
  bsum[0] = Wsum[0] + HIDDIM * HIDDIM;
  Wsum[1] = bsum[0] + HIDDIM;
  bsum[1] = Wsum[1] + HIDDIM * HIDDIM;

  const int mmBlocks = (NODES + 127) / 128;

  // 1) per-relation inverse in-degree (topology only: compute once, reuse)
  zero_kernel<<<(3 * NODES + 255) / 256, 256, 0, stream>>>(invcnt, 3 * NODES);
  count_kernel<<<(3 * EDGES + 255) / 256, 256, 0, stream>>>(ei[0], ei[1], ei[2], invcnt);
  invert_kernel<<<(3 * NODES + 255) / 256, 256, 0, stream>>>(invcnt, 3 * NODES);

  // 2) fold x @ Wr_r^T over relations into a single matmul with summed weights
  wsum_kernel<<<1, 256, 0, stream>>>(Wr[0], bl[0], Wsum[0], bsum[0]);
  wsum_kernel<<<1, 256, 0, stream>>>(Wr[1], bl[1], Wsum[1], bsum[1]);

  // 3) x0 = relu(concat(emb) @ W0^T + b0)
  embed_mlp_kernel<<<mmBlocks, 256, 0, stream>>>(shape_ids, color_ids, semb, cemb,
                                                 W0, b0, xA);

  // 4) two RGCN-SAGE layers
  float* xin = xA;
  float* xout = xB;
  for (int l = 0; l < 2; ++l) {
    // self-path + summed biases (init)
    matmul64_kernel<<<mmBlocks, 256, 0, stream>>>(xin, nullptr, Wsum[l], bsum[l],
                                                  xout, 0);
    for (int r = 0; r < NRELS; ++r) {
      zero_kernel<<<((NODES * HIDDIM) + 255) / 256, 256, 0, stream>>>(
          agg, NODES * HIDDIM);
      scatter_kernel<<<(4 * EDGES + 255) / 256, 256, 0, stream>>>(xin, ei[r], agg);
      int mode = 1 | ((r == NRELS - 1) ? 2 : 0);  // accumulate; relu on last
      matmul64_kernel<<<mmBlocks, 256, 0, stream>>>(
          agg, invcnt + (size_t)r * NODES, Wl[l] + (size_t)r * HIDDIM * HIDDIM,
          nullptr, xout, mode);
    }
    float* t = xin; xin = xout; xout = t;
  }

  // 5) global mean pool + output head
  zero_kernel<<<((GRAPHS * HIDDIM + GRAPHS) + 255) / 256, 256, 0, stream>>>(
      pooled, GRAPHS * HIDDIM + GRAPHS);
  pool_kernel<<<(4 * NODES + 255) / 256, 256, 0, stream>>>(xin, batch, pooled, gcnt);
  out_kernel<<<(GRAPHS + 255) / 256, 256, 0, stream>>>(pooled, gcnt, Wout, bout, out);
}